// PhysioAwareMasker_18588618457362
// MI455X (gfx1250) — compile-verified
//
#include <hip/hip_runtime.h>
#include <stdint.h>

#define BATCH     16384
#define NCH       12
#define SEQ_LEN   250u
#define SPAN_LEN  50u
#define NUM_SPANS 3
#define ATTEMPTS  15u
#define HI_BOUND  200u   // exclusive randint upper bound (SEQ_LEN - SPAN_LEN)
#define MULT200   96u    // ((2^16 % 200)^2) % 200, JAX randint bias-reduction multiplier
#define QRS_LO    45u    // cand <= 45  <=> end <= QRS_START(95)
#define QRS_HIu   155u   // cand >= 155 <=> cand >= QRS_END

typedef float v4f __attribute__((ext_vector_type(4)));

struct U2 { uint32_t a, b; };

// JAX threefry2x32: 20 rounds, key schedule injected every 4 rounds.
#define TFR(r) { x0 += x1; x1 = (x1 << (r)) | (x1 >> (32 - (r))); x1 ^= x0; }
__host__ __device__ constexpr U2 tf2x32(uint32_t k0, uint32_t k1, uint32_t x0, uint32_t x1) {
  uint32_t ks2 = k0 ^ k1 ^ 0x1BD11BDAu;
  x0 += k0; x1 += k1;
  TFR(13) TFR(15) TFR(26) TFR(6)   x0 += k1;  x1 += ks2 + 1u;
  TFR(17) TFR(29) TFR(16) TFR(24)  x0 += ks2; x1 += k0  + 2u;
  TFR(13) TFR(15) TFR(26) TFR(6)   x0 += k0;  x1 += k1  + 3u;
  TFR(17) TFR(29) TFR(16) TFR(24)  x0 += k1;  x1 += ks2 + 4u;
  TFR(13) TFR(15) TFR(26) TFR(6)   x0 += ks2; x1 += k0  + 5u;
  return U2{x0, x1};
}
#undef TFR

// ---- compile-time key derivation: key(42) -> split(3) -> randint sub-splits ----
// jax.random.key(42) == (0, 42). split(key, n): counts=iota(2n), pairs (i, i+n),
// out = [x0_0..x0_{n-1}, x1_0..x1_{n-1}], reshape (n,2).
constexpr U2 SP0 = tf2x32(0u, 42u, 0u, 3u);
constexpr U2 SP1 = tf2x32(0u, 42u, 1u, 4u);
constexpr U2 SP2 = tf2x32(0u, 42u, 2u, 5u);
constexpr uint32_t K1_0 = SP0.a, K1_1 = SP1.a;   // uniform key
constexpr uint32_t K2_0 = SP2.a, K2_1 = SP0.b;   // candidate randint key
constexpr uint32_t K3_0 = SP1.b, K3_1 = SP2.b;   // fallback randint key
// randint internally splits its key in two: counts=iota(4), pairs (0,2),(1,3)
constexpr U2 Q0 = tf2x32(K2_0, K2_1, 0u, 2u);
constexpr U2 Q1 = tf2x32(K2_0, K2_1, 1u, 3u);
constexpr uint32_t K2A_0 = Q0.a, K2A_1 = Q1.a;   // higher bits
constexpr uint32_t K2B_0 = Q0.b, K2B_1 = Q1.b;   // lower bits
constexpr U2 R0 = tf2x32(K3_0, K3_1, 0u, 2u);
constexpr U2 R1 = tf2x32(K3_0, K3_1, 1u, 3u);
constexpr uint32_t K3A_0 = R0.a, K3A_1 = R1.a;
constexpr uint32_t K3B_0 = R0.b, K3B_1 = R1.b;

// random_bits element i of an even-sized draw: pair (i, i+half) -> x0 for the
// first half of outputs, x1 for the second half.
__device__ __forceinline__ uint32_t rbits(uint32_t k0, uint32_t k1, uint32_t i, uint32_t half) {
  if (i < half) { U2 t = tf2x32(k0, k1, i, i + half); return t.a; }
  U2 t = tf2x32(k0, k1, i - half, i); return t.b;
}

// JAX randint in [0,200): ((hi % span) * (2^32 % span) + lo % span) % span
__device__ __forceinline__ uint32_t draw200(uint32_t ka0, uint32_t ka1,
                                            uint32_t kb0, uint32_t kb1,
                                            uint32_t i, uint32_t half) {
  uint32_t hb = rbits(ka0, ka1, i, half);
  uint32_t lb = rbits(kb0, kb1, i, half);
  return ((hb % HI_BOUND) * MULT200 + (lb % HI_BOUND)) % HI_BOUND;
}

// Kernel 1: one thread per batch -> 3 span starts (rejection sampling, 15 attempts)
__global__ __launch_bounds__(256) void spans_kernel(int* __restrict__ starts) {
  uint32_t b = blockIdx.x * 256u + threadIdx.x;
  if (b >= BATCH) return;
#pragma unroll
  for (uint32_t s = 0; s < NUM_SPANS; ++s) {
    uint32_t i = b * 3u + s;
    // uniform(k1, (B,3)): size 49152, half 24576; u = bitcast((bits>>9)|0x3F800000) - 1
    uint32_t ub = rbits(K1_0, K1_1, i, 24576u);
    float u = __uint_as_float((ub >> 9) | 0x3F800000u) - 1.0f;
    bool physio = u < 0.8f;
    int chosen = -1;
    for (uint32_t a = 0; a < ATTEMPTS; ++a) {
      uint32_t j = i * ATTEMPTS + a;                        // size 737280, half 368640
      uint32_t cand = draw200(K2A_0, K2A_1, K2B_0, K2B_1, j, 368640u);
      bool no_ov = (cand <= QRS_LO) || (cand >= QRS_HIu);
      if (chosen < 0 && (no_ov || !physio)) chosen = (int)cand;
    }
    if (chosen < 0)                                          // size 49152, half 24576
      chosen = (int)draw200(K3A_0, K3A_1, K3B_0, K3B_1, i, 24576u);
    starts[i] = chosen;
  }
}

// Kernel 2: masks output (B, L) as 0.0/1.0 floats, coalesced on L
__global__ __launch_bounds__(256) void mask_kernel(const int* __restrict__ starts,
                                                   float* __restrict__ maskf) {
  uint32_t t = blockIdx.x * 256u + threadIdx.x;
  if (t >= (uint32_t)BATCH * SEQ_LEN) return;
  uint32_t b = t / SEQ_LEN;
  uint32_t l = t - b * SEQ_LEN;
  uint32_t s0 = (uint32_t)starts[3u * b + 0];
  uint32_t s1 = (uint32_t)starts[3u * b + 1];
  uint32_t s2 = (uint32_t)starts[3u * b + 2];
  bool m = ((l - s0) < SPAN_LEN) | ((l - s1) < SPAN_LEN) | ((l - s2) < SPAN_LEN);
  maskf[t] = m ? 1.0f : 0.0f;
}

// Kernel 3: masked_x = x * keep. 3000 floats/batch == 750 float4s: chunks never
// straddle batches. Non-temporal 128-bit streaming; keep-predicate recomputed
// from 3 cached starts per batch (no 16MB mask re-read).
__global__ __launch_bounds__(256) void apply_kernel(const float* __restrict__ x,
                                                    const int* __restrict__ starts,
                                                    float* __restrict__ out) {
  const uint32_t n4 = (uint32_t)BATCH * NCH * SEQ_LEN / 4u;  // 12,288,000
  uint32_t t = blockIdx.x * 256u + threadIdx.x;
  if (t >= n4) return;
  uint32_t b = t / 750u;
  uint32_t l = ((t - b * 750u) * 4u) % SEQ_LEN;
  uint32_t s0 = (uint32_t)starts[3u * b + 0];
  uint32_t s1 = (uint32_t)starts[3u * b + 1];
  uint32_t s2 = (uint32_t)starts[3u * b + 2];
  v4f v = __builtin_nontemporal_load((const v4f*)x + t);
#pragma unroll
  for (int k = 0; k < 4; ++k) {
    bool m = ((l - s0) < SPAN_LEN) | ((l - s1) < SPAN_LEN) | ((l - s2) < SPAN_LEN);
    if (m) v[k] = 0.0f;
    l = (l + 1u == SEQ_LEN) ? 0u : l + 1u;
  }
  __builtin_nontemporal_store(v, (v4f*)out + t);
}

extern "C" void kernel_launch(void* const* d_in, const int* in_sizes, int n_in,
                              void* d_out, int out_size, void* d_ws, size_t ws_size,
                              hipStream_t stream) {
  (void)in_sizes; (void)n_in; (void)out_size; (void)ws_size;
  const float* x   = (const float*)d_in[0];
  float* out       = (float*)d_out;                                  // (B, C, L)
  float* maskf     = out + (size_t)BATCH * NCH * SEQ_LEN;            // (B, L)
  int*   starts    = (int*)d_ws;                                     // (B, 3) ints

  spans_kernel<<<(BATCH + 255) / 256, 256, 0, stream>>>(starts);
  mask_kernel<<<((uint32_t)BATCH * SEQ_LEN + 255u) / 256u, 256, 0, stream>>>(starts, maskf);
  apply_kernel<<<((uint32_t)BATCH * NCH * SEQ_LEN / 4u + 255u) / 256u, 256, 0, stream>>>(x, starts, out);
}